// CenterNet_85727547228596
// MI455X (gfx1250) — compile-verified
//
#include <hip/hip_runtime.h>
#include <hip/hip_bf16.h>
#include <stdint.h>

// CenterNet decode for MI455X (gfx1250): bandwidth-bound; TDM/async-LDS path.
#define B_      32
#define C_      80
#define H_      128
#define W_      128
#define HW_     (H_ * W_)
#define NCH_    84
#define K_      100
#define TILE_H  16
#define HALO    3
#define LROWS   (TILE_H + 2 * HALO)   // 22 rows incl. halo
#define LPAD    4                     // left/right -inf pad (16B-aligned dests)
#define LSTR    (W_ + 2 * LPAD)       // 136-float LDS row stride

#define NEG_INF (-__builtin_inff())

#ifndef USE_TDM
#if defined(__has_builtin)
#if __has_builtin(__builtin_amdgcn_tensor_load_to_lds)
#define USE_TDM 1
#endif
#endif
#endif
#ifndef USE_TDM
#define USE_TDM 0
#endif

#if USE_TDM
typedef unsigned int u32x4 __attribute__((ext_vector_type(4)));
typedef int          i32x8 __attribute__((ext_vector_type(8)));
typedef int          i32x4 __attribute__((ext_vector_type(4)));

// One-shot TDM DMA: ndw contiguous DWORDs from global -> LDS, with 8-DWORD
// LDS padding inserted after every 128 DWORDs (row stride 128 -> 136).
static __device__ inline void tdm_issue(const float* gsrc, uint32_t ldsDstByte,
                                        int ndw)
{
    uint64_t ga = (uint64_t)(size_t)gsrc;
    u32x4 g0;
    g0.x = 1u;                                   // count=1 (valid descriptor)
    g0.y = ldsDstByte;                           // lds_addr
    g0.z = (uint32_t)ga;                         // global_addr[31:0]
    g0.w = (uint32_t)((ga >> 32) & 0x1FFFFFFu)   // global_addr[56:32]
         | (2u << 30);                           // type=2 ("image")
    uint32_t dim0 = (uint32_t)ndw;               // elements (data_size units)
    i32x8 g1;
    g1[0] = (int)((2u << 16)                     // data_size = 4B
                | (1u << 20)                     // pad_enable
                | (6u << 22)                     // pad_interval: 128 DWORDs
                | (7u << 25));                   // pad_amount:   8 DWORDs
    g1[1] = (int)((dim0 & 0xFFFFu) << 16);       // tensor_dim0[15:0]
    g1[2] = (int)(dim0 >> 16);                   // tensor_dim0[31:16]
    g1[3] = (int)((dim0 & 0xFFFFu) << 16);       // tile_dim0 (1-D tile)
    g1[4] = 0;                                   // tile_dim1=0, tile_dim2=0
    g1[5] = (int)dim0;                           // tensor_dim0_stride (unused)
    g1[6] = 0;
    g1[7] = 0;
    i32x4 z4 = (i32x4)0;
    i32x8 z8 = (i32x8)0;
    // clang-23 / therock-10.0 6-arg form: (g0, g1, g2, g3, extra, cpol)
    __builtin_amdgcn_tensor_load_to_lds(g0, g1, z4, z4, z8, 0);
}

#if defined(__has_builtin) && __has_builtin(__builtin_amdgcn_s_wait_tensorcnt)
#define WAIT_TENSOR_1() __builtin_amdgcn_s_wait_tensorcnt(1)
#define WAIT_TENSOR_0() __builtin_amdgcn_s_wait_tensorcnt(0)
#else
#define WAIT_TENSOR_1() asm volatile("s_wait_tensorcnt 1" ::: "memory")
#define WAIT_TENSOR_0() asm volatile("s_wait_tensorcnt 0" ::: "memory")
#endif
#endif // USE_TDM

// ---------------------------------------------------------------------------
// Pass 1: per (batch, 16-row tile): for each of 80 class planes, DMA a 22x128
// haloed logit tile into LDS (TDM with row padding, or per-wave async b128),
// separable 7x7 max-pool, equality NMS mask, running channel max (logits).
// ---------------------------------------------------------------------------
__global__ __launch_bounds__(256)
void centernet_heat_kernel(const float* __restrict__ in,
                           float* __restrict__ scores,
                           float* __restrict__ clsArr)
{
    __shared__ __align__(16) float lbuf[2][LROWS][LSTR];  // padded logit tiles
    __shared__ __align__(16) float hbuf[LROWS][W_];       // horizontal max
    __shared__ __align__(16) float dump[4];               // async idle-lane sink

    const int tid   = threadIdx.x;
    const int wid   = tid >> 5;
    const int tileY = blockIdx.x;            // 0..7
    const int b     = blockIdx.y;            // 0..31
    const int gy0   = tileY * TILE_H - HALO; // first halo row (may be <0)
    const int r0    = (gy0 < 0) ? -gy0 : 0;                       // first valid lrow
    const int r1    = (gy0 + LROWS > H_) ? (H_ - gy0) : LROWS;    // past-last valid

    // Pre-fill pad columns and out-of-image halo rows with -inf. These cells
    // are disjoint from every DMA destination, so no ordering is needed.
    for (int r = 0; r < LROWS; ++r) {
        bool rowInvalid = (r < r0) || (r >= r1);
        for (int x = tid; x < LSTR; x += 256) {
            if (rowInvalid || (x < LPAD) || (x >= LPAD + W_)) {
                lbuf[0][r][x] = NEG_INF;
                lbuf[1][r][x] = NEG_INF;
            }
        }
    }

    const float* planes = in + (size_t)(b * NCH_ + 4) * HW_;  // class logits

#if USE_TDM
    const int nrows = r1 - r0;                 // valid rows: contiguous source
    auto issue = [&](int c, int bi) {          // call under (wid == 0)
        const float* gsrc = planes + (size_t)c * HW_ + (size_t)(gy0 + r0) * W_;
        uint32_t dst = (uint32_t)(size_t)&lbuf[bi][r0][LPAD];
        tdm_issue(gsrc, dst, nrows * W_);
    };
    if (wid == 0) { issue(0, 0); issue(1, 1); }
#else
    const uint32_t dumpOff = (uint32_t)(size_t)&dump[0];
    auto issue = [&](int c, int bi) {          // all waves: 3 async instrs each
        const float* gbase = planes + (size_t)c * HW_;
        float* lbase = &lbuf[bi][0][0];
#pragma unroll
        for (int i = 0; i < 3; ++i) {
            int s   = tid + i * 256;           // slot id 0..767
            int row = s >> 5;                  // 32 b128 quads per row
            int x0  = (s & 31) << 2;
            bool valid = (row >= r0) && (row < r1);
            const float* g = gbase + (valid ? ((gy0 + row) * W_ + x0) : 0);
            uint32_t loff  = valid ? (uint32_t)(size_t)(lbase + row * LSTR + LPAD + x0)
                                   : dumpOff;
            asm volatile("global_load_async_to_lds_b128 %0, %1, off"
                         :: "v"(loff), "v"(g) : "memory");
        }
    };
    issue(0, 0);
    issue(1, 1);
#endif

    float best[8];
    int   bestc[8];
#pragma unroll
    for (int i = 0; i < 8; ++i) { best[i] = NEG_INF; bestc[i] = 0; }

    for (int c = 0; c < C_; ++c) {
        const int cur = c & 1;
#if USE_TDM
        if (wid == 0) {
            if (c + 1 < C_) WAIT_TENSOR_1();
            else            WAIT_TENSOR_0();
        }
#else
        if (c + 1 < C_) asm volatile("s_wait_asynccnt 3" ::: "memory");
        else            asm volatile("s_wait_asynccnt 0" ::: "memory");
#endif
        __syncthreads();   // tile visible to all waves; hbuf free to overwrite

        // Horizontal 7-wide max, branch-free via -inf column pads.
#pragma unroll
        for (int i = 0; i < 11; ++i) {         // 22*128 entries, 11/thread
            int e = tid + i * 256;
            int r = e >> 7;
            int x = e & (W_ - 1);
            float m = lbuf[cur][r][x + 1];     // padded cols (x+4)-3 .. (x+4)+3
#pragma unroll
            for (int dx = 2; dx <= 7; ++dx) m = fmaxf(m, lbuf[cur][r][x + dx]);
            hbuf[r][x] = m;
        }
        __syncthreads();   // hbuf ready

        // Vertical 7-tall max + equality NMS + running channel max (logits).
#pragma unroll
        for (int i = 0; i < 8; ++i) {
            int p = tid + i * 256;
            int y = p >> 7;                    // 0..15
            int x = p & (W_ - 1);
            float center = lbuf[cur][y + HALO][x + LPAD];
            float m = hbuf[y][x];
#pragma unroll
            for (int dy = 1; dy < 7; ++dy) m = fmaxf(m, hbuf[y + dy][x]);
            if ((center == m) && (center > best[i])) { best[i] = center; bestc[i] = c; }
        }
        __syncthreads();   // all reads of lbuf[cur]/hbuf done -> safe to refill

        if (c + 2 < C_) {
#if USE_TDM
            if (wid == 0) issue(c + 2, cur);
#else
            issue(c + 2, cur);
#endif
        }
    }

    // Emit sigmoid(best logit) and class (sigmoid(-inf)=0 => "no max" case).
#pragma unroll
    for (int i = 0; i < 8; ++i) {
        int p  = tid + i * 256;
        int y  = p >> 7;
        int x  = p & (W_ - 1);
        int gy = tileY * TILE_H + y;
        size_t o = (size_t)b * HW_ + (size_t)gy * W_ + x;
        scores[o] = 1.0f / (1.0f + __expf(-best[i]));
        clsArr[o] = (float)bestc[i];
    }
    (void)dump;
}

// ---------------------------------------------------------------------------
// Pass 2: per-batch iterative top-100 (wave32 shfl_xor + LDS cross-wave
// reduce, lowest-index tie-break), then gather bbox regressors + emit dets.
// ---------------------------------------------------------------------------
__global__ __launch_bounds__(1024)
void centernet_topk_kernel(const float* __restrict__ in,
                           const float* __restrict__ scores,
                           const float* __restrict__ clsArr,
                           float* __restrict__ dets)
{
    const int tid  = threadIdx.x;
    const int b    = blockIdx.x;
    const int lane = tid & 31;
    const int wid  = tid >> 5;

    __shared__ float wv[32];
    __shared__ int   wi[32];
    __shared__ float kv[K_];
    __shared__ int   ki[K_];
    __shared__ float bV;
    __shared__ int   bI;

    float lv[16];
    const float* sb = scores + (size_t)b * HW_;
#pragma unroll
    for (int i = 0; i < 16; ++i) lv[i] = sb[tid + i * 1024];

    float lmax; int lidx;
    auto rescan = [&]() {
        lmax = lv[0]; lidx = tid;
#pragma unroll
        for (int i = 1; i < 16; ++i)
            if (lv[i] > lmax) { lmax = lv[i]; lidx = tid + i * 1024; }
    };
    rescan();

    for (int k = 0; k < K_; ++k) {
        float v = lmax; int idx = lidx;
#pragma unroll
        for (int m = 16; m >= 1; m >>= 1) {
            float ov = __shfl_xor(v, m);
            int   oi = __shfl_xor(idx, m);
            if (ov > v || (ov == v && oi < idx)) { v = ov; idx = oi; }
        }
        if (lane == 0) { wv[wid] = v; wi[wid] = idx; }
        __syncthreads();
        if (tid < 32) {
            v = wv[tid]; idx = wi[tid];
#pragma unroll
            for (int m = 16; m >= 1; m >>= 1) {
                float ov = __shfl_xor(v, m);
                int   oi = __shfl_xor(idx, m);
                if (ov > v || (ov == v && oi < idx)) { v = ov; idx = oi; }
            }
            if (tid == 0) { bV = v; bI = idx; kv[k] = v; ki[k] = idx; }
        }
        __syncthreads();
        int widx = bI;
        if ((widx & 1023) == tid) {     // owner removes winner, rescans locals
            lv[widx >> 10] = NEG_INF;
            rescan();
        }
        // next iteration's barrier (after wv write) orders bV/bI reuse
    }

    __syncthreads();
    if (tid < K_) {
        int   idx   = ki[tid];
        float score = kv[tid];
        int y = idx >> 7;
        int x = idx & (W_ - 1);
        const float* pb = in + (size_t)b * NCH_ * HW_ + idx;
        float tx = pb[0 * HW_], ty = pb[1 * HW_];
        float tw = pb[2 * HW_], th = pb[3 * HW_];
        const float inv = 1.0f / 128.0f;   // STRIDE/img = 4/512
        float* o = dets + ((size_t)b * K_ + tid) * 6;
        o[0] = clsArr[(size_t)b * HW_ + idx];
        o[1] = ((float)x + tx) * inv;
        o[2] = ((float)y + ty) * inv;
        o[3] = tw * inv;
        o[4] = th * inv;
        o[5] = score;
    }
}

extern "C" void kernel_launch(void* const* d_in, const int* in_sizes, int n_in,
                              void* d_out, int out_size, void* d_ws, size_t ws_size,
                              hipStream_t stream) {
    const float* in = (const float*)d_in[0];      // [32, 84, 128, 128] f32
    float* dets     = (float*)d_out;              // [32, 100, 6] f32
    float* scores   = (float*)d_ws;               // [32, 16384]
    float* clsArr   = scores + (size_t)B_ * HW_;  // [32, 16384]

    dim3 g1(H_ / TILE_H, B_);                     // 8 x 32 = 256 workgroups
    centernet_heat_kernel<<<g1, 256, 0, stream>>>(in, scores, clsArr);
    centernet_topk_kernel<<<B_, 1024, 0, stream>>>(in, scores, clsArr, dets);
}